// ContinuousDeltaModel_3204045603492
// MI455X (gfx1250) — compile-verified
//
#include <hip/hip_runtime.h>

typedef __attribute__((ext_vector_type(2))) float v2f;
typedef __attribute__((ext_vector_type(8))) float v8f;

#define HD 32
#define VD 64
#define TCH 64           // tokens per chunk
#define NTHREADS 160     // 5 waves: waves 0-3 encode (16 rows each), wave 4 scans

// LDS row strides (floats):
//  s_h : 34 -> byte stride 136 (mult of 8: aligned b64 A-frag loads, conflict-free)
//  s_ff: 66 -> byte stride 264 (mult of 8)
//  s_x : 36 -> byte stride 144 (mult of 16: aligned float4 broadcast loads in scan);
//              col 32 stores precomputed rinv = 1/(k.k + 1e-6)

__launch_bounds__(NTHREADS)
__global__ void delta_model_fused(
    const int*   __restrict__ seq,
    const float* __restrict__ embed,
    const float* __restrict__ w1, const float* __restrict__ b1,
    const float* __restrict__ w2, const float* __restrict__ b2,
    const float* __restrict__ ln_g, const float* __restrict__ ln_b,
    const float* __restrict__ wr, const float* __restrict__ br,
    const float* __restrict__ wo, const float* __restrict__ bo,
    float* __restrict__ out, int L)
{
  __shared__ float s_w1[HD * 2 * HD];    // 32x64
  __shared__ float s_w2[2 * HD * HD];    // 64x32
  __shared__ float s_b1[2 * HD];
  __shared__ float s_b2[HD];
  __shared__ float s_g[HD];
  __shared__ float s_bt[HD];
  __shared__ float s_h [TCH * 34];
  __shared__ float s_ff[TCH * 66];
  __shared__ __align__(16) float s_x[2][TCH * 36];
  __shared__ float s_ctx[HD];
  __shared__ float s_tmp[HD];

  const int tid  = threadIdx.x;
  const int wave = __builtin_amdgcn_readfirstlane(tid) >> 5;   // SGPR-uniform
  const int lane = tid & 31;
  const int ln   = lane & 15;   // N / M index within tile
  const int hl   = lane >> 4;   // selects K pair (A/B) / rows M+8 (C)
  const int bidx = blockIdx.x;

  // ---- preload weights into LDS ----
  for (int i = tid; i < HD * 2 * HD; i += NTHREADS) s_w1[i] = w1[i];
  for (int i = tid; i < 2 * HD * HD; i += NTHREADS) s_w2[i] = w2[i];
  if (tid < 2 * HD) s_b1[tid] = b1[tid];
  if (tid < HD) { s_b2[tid] = b2[tid]; s_g[tid] = ln_g[tid]; s_bt[tid] = ln_b[tid]; }

  // Row of the delta-memory M per lane (scan wave only uses it).
  float Mrow[HD];
#pragma unroll
  for (int j = 0; j < HD; ++j) Mrow[j] = 0.f;

  const int nChunks = L / TCH;
  const int m0 = (wave & 3) * 16;   // encode wave's 16-row slab

  // Self-contained encode of one chunk slab by one wave (no cross-wave deps).
  auto encode = [&](int cIdx, float* xb) {
    const int t0 = cIdx * TCH;
    // gather: 16 rows of h = embed[seq]; row per iteration, lanes = columns
#pragma unroll
    for (int r = 0; r < 16; ++r) {
      int tok = seq[(size_t)bidx * L + t0 + m0 + r];
      s_h[(m0 + r) * 34 + lane] = embed[tok * HD + lane];
    }
    // GEMM1: relu(h @ w1 + b1) -> s_ff slab (16 x 64)
#pragma unroll
    for (int nt = 0; nt < 4; ++nt) {
      const int n0 = nt * 16;
      v8f acc = {0.f, 0.f, 0.f, 0.f, 0.f, 0.f, 0.f, 0.f};
#pragma unroll
      for (int k0 = 0; k0 < HD; k0 += 4) {
        const float* ap = &s_h[(m0 + ln) * 34 + k0 + 2 * hl];
        v2f a; a.x = ap[0]; a.y = ap[1];
        v2f bf; bf.x = s_w1[(k0 + 2 * hl) * 64 + n0 + ln];
        bf.y = s_w1[(k0 + 2 * hl + 1) * 64 + n0 + ln];
        acc = __builtin_amdgcn_wmma_f32_16x16x4_f32(false, a, false, bf,
                                                    (short)0, acc, false, false);
      }
      float bias = s_b1[n0 + ln];
#pragma unroll
      for (int r = 0; r < 8; ++r) {
        int row = m0 + r + 8 * hl;
        float v = acc[r] + bias;
        s_ff[row * 66 + n0 + ln] = v > 0.f ? v : 0.f;
      }
    }
    // GEMM2: x = ff @ w2 + b2 + h -> xb slab (16 x 32)
#pragma unroll
    for (int nt = 0; nt < 2; ++nt) {
      const int n0 = nt * 16;
      v8f acc = {0.f, 0.f, 0.f, 0.f, 0.f, 0.f, 0.f, 0.f};
#pragma unroll
      for (int k0 = 0; k0 < 2 * HD; k0 += 4) {
        const float* ap = &s_ff[(m0 + ln) * 66 + k0 + 2 * hl];
        v2f a; a.x = ap[0]; a.y = ap[1];
        v2f bf; bf.x = s_w2[(k0 + 2 * hl) * HD + n0 + ln];
        bf.y = s_w2[(k0 + 2 * hl + 1) * HD + n0 + ln];
        acc = __builtin_amdgcn_wmma_f32_16x16x4_f32(false, a, false, bf,
                                                    (short)0, acc, false, false);
      }
      float bias = s_b2[n0 + ln];
#pragma unroll
      for (int r = 0; r < 8; ++r) {
        int row = m0 + r + 8 * hl, col = n0 + ln;
        xb[row * 36 + col] = acc[r] + bias + s_h[row * 34 + col];
      }
    }
    // LayerNorm + rinv precompute: lane l < 16 owns row m0+l (same-wave data)
    if (lane < 16) {
      float* xr = &xb[(m0 + lane) * 36];
      float mu = 0.f;
#pragma unroll
      for (int j = 0; j < HD; ++j) mu += xr[j];
      mu *= (1.f / 32.f);
      float var = 0.f;
#pragma unroll
      for (int j = 0; j < HD; ++j) { float d = xr[j] - mu; var = fmaf(d, d, var); }
      var *= (1.f / 32.f);
      float rs = __builtin_amdgcn_rsqf(var + 1e-5f);
      float kk2 = 0.f;
#pragma unroll
      for (int j = 0; j < HD; ++j) {
        float v = (xr[j] - mu) * rs * s_g[j] + s_bt[j];
        xr[j] = v;
        kk2 = fmaf(v, v, kk2);
      }
      xr[HD] = __builtin_amdgcn_rcpf(kk2 + 1e-6f);   // rinv in spare col 32
    }
  };

  __syncthreads();                 // weights visible
  if (wave < 4) encode(0, s_x[0]); // prologue: chunk 0 -> buf 0
  __syncthreads();

  // Pipelined main loop: wave 4 scans chunk c while waves 0-3 encode chunk c+1.
  for (int c = 0; c < nChunks; ++c) {
    float* xb_cur = s_x[c & 1];
    float* xb_nxt = s_x[(c & 1) ^ 1];

    if (wave < 4) {
      if (c + 1 < nChunks) encode(c + 1, xb_nxt);
    } else {
      const bool last = (c == nChunks - 1);
      const int steps = last ? (TCH - 1) : TCH;    // scan covers t = 0 .. L-2

      // issue step t's LDS broadcast loads
      auto loadk = [&](int t, float (&kkv)[HD], float& kl, float& ri) {
        const float4* kp = (const float4*)&xb_cur[t * 36];
#pragma unroll
        for (int q = 0; q < 8; ++q) {
          float4 f = kp[q];
          kkv[4 * q] = f.x; kkv[4 * q + 1] = f.y;
          kkv[4 * q + 2] = f.z; kkv[4 * q + 3] = f.w;
        }
        ri = xb_cur[t * 36 + HD];
        kl = xb_cur[t * 36 + lane];
      };
      // one delta-rule step from registers
      auto stepM = [&](const float (&kkv)[HD], float kl, float ri) {
        float vp0 = 0.f, vp1 = 0.f, vp2 = 0.f, vp3 = 0.f;
#pragma unroll
        for (int j = 0; j < HD; j += 4) {
          vp0 = fmaf(Mrow[j],     kkv[j],     vp0);
          vp1 = fmaf(Mrow[j + 1], kkv[j + 1], vp1);
          vp2 = fmaf(Mrow[j + 2], kkv[j + 2], vp2);
          vp3 = fmaf(Mrow[j + 3], kkv[j + 3], vp3);
        }
        float vp = (vp0 + vp1) + (vp2 + vp3);
        float u = fmaf(-vp, ri, kl);
#pragma unroll
        for (int j = 0; j < HD; ++j) Mrow[j] = fmaf(u, kkv[j], Mrow[j]);
      };

      // software-pipelined, unroll-by-2 with ping-pong register buffers
      float kkA[HD], kkB[HD];
      float klA, riA, klB, riB;
      loadk(0, kkA, klA, riA);
      for (int t = 0; t < steps; t += 2) {
        if (t + 1 < steps) loadk(t + 1, kkB, klB, riB);   // prefetch odd step
        stepM(kkA, klA, riA);
        if (t + 2 < steps) loadk(t + 2, kkA, klA, riA);   // prefetch next even
        if (t + 1 < steps) stepM(kkB, klB, riB);
      }

      if (last) {
        // ctx[i] = sum_j M[i][j] * hidden[L-1][j]
        const float* kl = &xb_cur[(TCH - 1) * 36];
        float c0 = 0.f, c1 = 0.f, c2 = 0.f, c3 = 0.f;
#pragma unroll
        for (int j = 0; j < HD; j += 4) {
          c0 = fmaf(Mrow[j],     kl[j],     c0);
          c1 = fmaf(Mrow[j + 1], kl[j + 1], c1);
          c2 = fmaf(Mrow[j + 2], kl[j + 2], c2);
          c3 = fmaf(Mrow[j + 3], kl[j + 3], c3);
        }
        s_ctx[lane] = (c0 + c1) + (c2 + c3);
      }
    }
    __syncthreads();
  }

  // ---- head: (ctx @ wr + br) @ wo + bo ----
  if (tid < HD) {
    float acc = br[tid];
#pragma unroll
    for (int j = 0; j < HD; ++j) acc = fmaf(s_ctx[j], wr[j * HD + tid], acc);
    s_tmp[tid] = acc;
  }
  __syncthreads();
  if (tid < VD) {
    float acc = bo[tid];
#pragma unroll
    for (int i = 0; i < HD; ++i) acc = fmaf(s_tmp[i], wo[i * VD + tid], acc);
    out[(size_t)bidx * VD + tid] = acc;
  }
}

extern "C" void kernel_launch(void* const* d_in, const int* in_sizes, int n_in,
                              void* d_out, int out_size, void* d_ws, size_t ws_size,
                              hipStream_t stream) {
  const int*   seq   = (const int*)  d_in[0];
  const float* embed = (const float*)d_in[1];
  const float* w1    = (const float*)d_in[2];
  const float* b1    = (const float*)d_in[3];
  const float* w2    = (const float*)d_in[4];
  const float* b2    = (const float*)d_in[5];
  const float* ln_g  = (const float*)d_in[6];
  const float* ln_b  = (const float*)d_in[7];
  const float* wr    = (const float*)d_in[8];
  const float* br    = (const float*)d_in[9];
  const float* wo    = (const float*)d_in[10];
  const float* bo    = (const float*)d_in[11];
  float* out = (float*)d_out;

  const int B = out_size / VD;          // 512
  const int L = in_sizes[0] / B;        // 2048

  delta_model_fused<<<B, NTHREADS, 0, stream>>>(
      seq, embed, w1, b1, w2, b2, ln_g, ln_b, wr, br, wo, bo, out, L);
}